// _QuantumGate_65481071409233
// MI455X (gfx1250) — compile-verified
//
#include <hip/hip_runtime.h>

typedef float v2f __attribute__((ext_vector_type(2)));
typedef float v8f __attribute__((ext_vector_type(8)));

#define NQ     10
#define WAVES  16
#define BLOCK  (WAVES * 32)
#define ROW    1024                  // floats per probs row in LDS
#define SMEM_FLOATS (WAVES * ROW)    // 16384 floats = 64 KB

// XOR swizzle (multiple of 4 floats) keeps (k,k+1) pairs adjacent while
// making the 16 rows hit distinct banks for the WMMA A-matrix gather.
__device__ __forceinline__ int swz(int row, int k) {
  return row * ROW + (k ^ ((row & 15) << 2));
}

__global__ __launch_bounds__(BLOCK)
void qsim10(const float* __restrict__ x, const float* __restrict__ p,
            float* __restrict__ out, int nbatch) {
  extern __shared__ float smem[];
  const int lane = threadIdx.x & 31;
  const int wave = threadIdx.x >> 5;
  const int batch = blockIdx.x * WAVES + wave;

  // Full 1024-amplitude state in registers: k = (j<<5) | lane
  // (bits 9..5 = register index j, bits 4..0 = lane)
  float re[32], im[32];
#pragma unroll
  for (int j = 0; j < 32; ++j) { re[j] = 0.f; im[j] = 0.f; }
  if (lane == 0) re[0] = 1.f;

  // ---------------- RX(x[b][i]) on wire i (all RX commute) --------------
  // wires 0..4 -> qubit bit 9-i -> j bit (4-i): in-register pairs
#pragma unroll
  for (int i = 0; i < 5; ++i) {
    const float th = 0.5f * x[batch * NQ + i];
    const float c = __builtin_cosf(th), s = __builtin_sinf(th);
    const int bit = 4 - i;
#pragma unroll
    for (int j0 = 0; j0 < 32; ++j0) {
      if (j0 & (1 << bit)) continue;
      const int j1 = j0 | (1 << bit);
      const float r0 = re[j0], i0 = im[j0], r1 = re[j1], i1 = im[j1];
      re[j0] = c * r0 + s * i1;  im[j0] = c * i0 - s * r1;
      re[j1] = c * r1 + s * i0;  im[j1] = c * i1 - s * r0;
    }
  }
  // wires 5..9 -> qubit bit (9-i) in 4..0 -> lane bit: cross-lane shuffle
#pragma unroll
  for (int i = 5; i < 10; ++i) {
    const float th = 0.5f * x[batch * NQ + i];
    const float c = __builtin_cosf(th), s = __builtin_sinf(th);
    const int mask = 1 << (9 - i);
#pragma unroll
    for (int j = 0; j < 32; ++j) {
      const float pre = __shfl_xor(re[j], mask, 32);
      const float pim = __shfl_xor(im[j], mask, 32);
      re[j] = c * re[j] + s * pim;   // same formula for both pair halves
      im[j] = c * im[j] - s * pre;
    }
  }

  // ---------------- RY(p[i]) on wire i ----------------------------------
#pragma unroll
  for (int i = 0; i < 5; ++i) {
    const float th = 0.5f * p[i];
    const float c = __builtin_cosf(th), s = __builtin_sinf(th);
    const int bit = 4 - i;
#pragma unroll
    for (int j0 = 0; j0 < 32; ++j0) {
      if (j0 & (1 << bit)) continue;
      const int j1 = j0 | (1 << bit);
      const float r0 = re[j0], i0 = im[j0], r1 = re[j1], i1 = im[j1];
      re[j0] = c * r0 - s * r1;  im[j0] = c * i0 - s * i1;
      re[j1] = c * r1 + s * r0;  im[j1] = c * i1 + s * i0;
    }
  }
#pragma unroll
  for (int i = 5; i < 10; ++i) {
    const float th = 0.5f * p[i];
    const float c = __builtin_cosf(th), s = __builtin_sinf(th);
    const int mask = 1 << (9 - i);
    const float se = ((lane >> (9 - i)) & 1) ? s : -s;
#pragma unroll
    for (int j = 0; j < 32; ++j) {
      const float pre = __shfl_xor(re[j], mask, 32);
      const float pim = __shfl_xor(im[j], mask, 32);
      re[j] = c * re[j] + se * pre;
      im[j] = c * im[j] + se * pim;
    }
  }

  // ---------------- CNOT ladder (ctrl i, tgt i+1), i = 0..8 -------------
  // i=0..3: ctrl j-bit (4-i), tgt j-bit (3-i): pure register permutation
#pragma unroll
  for (int i = 0; i < 4; ++i) {
    const int bc = 4 - i, bt = 3 - i;
#pragma unroll
    for (int j = 0; j < 32; ++j) {
      if (((j >> bc) & 1) && !((j >> bt) & 1)) {
        const int jp = j | (1 << bt);
        float t = re[j]; re[j] = re[jp]; re[jp] = t;
        t = im[j]; im[j] = im[jp]; im[jp] = t;
      }
    }
  }
  // i=4: ctrl j-bit 0, tgt lane-bit 4: unconditional lane swap for odd j
#pragma unroll
  for (int j = 1; j < 32; j += 2) {
    re[j] = __shfl_xor(re[j], 16, 32);
    im[j] = __shfl_xor(im[j], 16, 32);
  }
  // i=5..8: ctrl lane-bit (9-i), tgt lane-bit (8-i): predicated lane swap
#pragma unroll
  for (int i = 5; i < 9; ++i) {
    const int mask = 1 << (8 - i);
    const bool cond = (lane >> (9 - i)) & 1;
#pragma unroll
    for (int j = 0; j < 32; ++j) {
      const float pr = __shfl_xor(re[j], mask, 32);
      const float pi = __shfl_xor(im[j], mask, 32);
      re[j] = cond ? pr : re[j];
      im[j] = cond ? pi : im[j];
    }
  }

  // ---------------- probabilities -> LDS --------------------------------
#pragma unroll
  for (int j = 0; j < 32; ++j) {
    const int k = (j << 5) | lane;        // lanes write consecutive addrs
    smem[swz(wave, k)] = re[j] * re[j] + im[j] * im[j];
  }
  __syncthreads();

  // ------- <Z_i> as GEMM via V_WMMA_F32_16X16X4_F32 ----------------------
  // D[16 batch x 16 wire] += A[16 x 4] * B[4 x 16], K = 1024 split over
  // 16 waves (64 each, 16 WMMA steps). B = sign matrix built in VALU.
  const int m = lane & 15;       // A row (batch in tile) == B col (wire)
  const int half = lane >> 4;    // ISA layout: lanes 16-31 carry K+2 pair
  const int shift = (m < NQ) ? (9 - m) : 0;
  v8f acc = {0.f, 0.f, 0.f, 0.f, 0.f, 0.f, 0.f, 0.f};
#pragma unroll
  for (int t = 0; t < 16; ++t) {
    const int k = wave * 64 + t * 4 + half * 2;
    v2f a, b;
    a.x = smem[swz(m, k)];
    a.y = smem[swz(m, k + 1)];
    b.x = (m < NQ) ? ((((k)     >> shift) & 1) ? -1.f : 1.f) : 0.f;
    b.y = (m < NQ) ? ((((k + 1) >> shift) & 1) ? -1.f : 1.f) : 0.f;
    acc = __builtin_amdgcn_wmma_f32_16x16x4_f32(false, a, false, b,
                                                (short)0, acc, false, false);
  }
  __syncthreads();   // all probs reads done; safe to reuse smem

  // partial C tiles -> LDS per ISA C/D layout (VGPR r: M=r / r+8 by half)
#pragma unroll
  for (int r = 0; r < 8; ++r) {
    const int mm = r + half * 8;
    smem[wave * 256 + mm * 16 + m] = acc[r];
  }
  __syncthreads();

  // reduce 16 partial tiles and emit out[b][i], i < 10
  if (threadIdx.x < 256) {
    const int mm = threadIdx.x >> 4, nn = threadIdx.x & 15;
    float ssum = 0.f;
#pragma unroll
    for (int w = 0; w < WAVES; ++w) ssum += smem[w * 256 + mm * 16 + nn];
    const int b = blockIdx.x * WAVES + mm;
    if (nn < NQ && b < nbatch) out[b * NQ + nn] = ssum;
  }
}

extern "C" void kernel_launch(void* const* d_in, const int* in_sizes, int n_in,
                              void* d_out, int out_size, void* d_ws, size_t ws_size,
                              hipStream_t stream) {
  (void)n_in; (void)d_ws; (void)ws_size; (void)out_size;
  const float* x = (const float*)d_in[0];       // [B, 10] f32
  const float* params = (const float*)d_in[1];  // [10] f32
  float* out = (float*)d_out;                   // [B, 10] f32
  const int nbatch = in_sizes[0] / NQ;
  const int grid = (nbatch + WAVES - 1) / WAVES;
  qsim10<<<grid, BLOCK, SMEM_FLOATS * sizeof(float), stream>>>(x, params, out, nbatch);
}